// mRNNLayer_17506286698942
// MI455X (gfx1250) — compile-verified
//
#include <hip/hip_runtime.h>
#include <cstdint>

#define H  1024
#define B  8
#define T  4096
#define C  (2 * H)

#define BH      128        // hidden columns per block
#define WCOLS   64         // hidden columns per wave (2 chains per lane)
#define TC      16         // timesteps per prefetch chunk
#define NSTAGE  3          // LDS ring stages per wave (2 chunks in flight)
#define NCHUNK  (T / TC)   // 256 chunks

// Per wave per chunk: 8 async b128 issues for xin + 8 for xf = 16 ops
// (each op: 32 lanes x 16B = 512B = 2 rows x 64 cols). Two chunks in
// flight -> ASYNCcnt <= 48 outstanding (6-bit counter max 63: OK).

// CDNA5 async global->LDS copy, tracked by ASYNCcnt.
#define ASYNC_LOAD_B128(ldsoff_u32, gaddr_u64)                              \
  asm volatile("global_load_async_to_lds_b128 %0, %1, off"                  \
               :: "v"(ldsoff_u32), "v"(gaddr_u64) : "memory")

#define WAIT_ASYNC(n) asm volatile("s_wait_asynccnt " #n ::: "memory")

// One recurrence step. gfx1250 has v_tanh_f32: sigmoid(z)=0.5+0.5*tanh(z/2)
// gives a 3-op dependent chain (fma -> tanh -> fma); the `u = s + xin/2`
// term runs in parallel with the tanh. Fallback: exp2/rcp (5-op chain).
__device__ __forceinline__ void step(float& s, float xin, float xf, float wh) {
#if __has_builtin(__builtin_amdgcn_tanhf)
  const float xi2 = 0.5f * xin;              // off-chain
  const float a   = __builtin_fmaf(s, wh, 0.5f * xf);
  const float th  = __builtin_amdgcn_tanhf(a);
  const float u   = s + xi2;                 // parallel with tanh
  s = __builtin_fmaf(th, xi2, u);
#else
  // wh here is the raw w ; -log2(e) scaling is applied in the fma below.
  const float c2 = -1.44269504088896340736f;
  const float a  = __builtin_fmaf(s, 2.0f * c2 * wh, xf * c2);
  const float e  = __builtin_amdgcn_exp2f(a);
  const float f  = __builtin_amdgcn_rcpf(1.0f + e);
  s = __builtin_fmaf(f, xin, s);
#endif
}

// Issue one chunk (TC timesteps x WCOLS h, xin+xf) of async loads for THIS
// wave only, into its private LDS region. lane l -> row=(l>>4)+2k, col=(l&15)*4
// => each issue is 2 rows x 16 lanes x 16B (256B contiguous per row: coalesced).
__device__ __forceinline__ void issue_chunk(const float* __restrict__ xbase,
                                            int t0,
                                            float (*buf)[TC][WCOLS], // [2][TC][WCOLS]
                                            int lane) {
  const int r2 = lane >> 4;          // 0..1
  const int c4 = (lane & 15) << 2;   // 0,4,...,60
#pragma unroll
  for (int k = 0; k < TC / 2; ++k) {
    const int row = (k << 1) + r2;
    const float* gi = xbase + (size_t)(t0 + row) * C + c4;  // xin
    const float* gf = gi + H;                               // xf
    const uint32_t li = (uint32_t)(uintptr_t)&buf[0][row][c4];
    const uint32_t lf = (uint32_t)(uintptr_t)&buf[1][row][c4];
    ASYNC_LOAD_B128(li, (uint64_t)(uintptr_t)gi);
    ASYNC_LOAD_B128(lf, (uint64_t)(uintptr_t)gf);
  }
}

__global__ __launch_bounds__(64) void mrnn_scan_kernel(
    const float* __restrict__ x,      // (B, T, 2H)
    const float* __restrict__ state,  // (B, H)
    const float* __restrict__ weight, // (H)
    float* __restrict__ out) {        // (B, T, H)
  // Per-wave private staging: no cross-wave LDS access -> NO barriers needed.
  __shared__ float lds[2][NSTAGE][2][TC][WCOLS];   // [wave][stage][in/f][t][col]

  const int tid  = threadIdx.x;
  const int wave = tid >> 5;
  const int lane = tid & 31;
  const int b    = blockIdx.y;
  const int h0   = blockIdx.x * BH + wave * WCOLS; // this wave's first column

  const float* xbase = x + (size_t)b * T * C + h0;
  float (*mylds)[2][TC][WCOLS] = lds[wave];

  // Two independent chains per lane: columns (lane) and (lane+32).
  float s0 = state[b * H + h0 + lane];
  float s1 = state[b * H + h0 + lane + 32];
#if __has_builtin(__builtin_amdgcn_tanhf)
  const float wh0 = 0.5f * weight[h0 + lane];
  const float wh1 = 0.5f * weight[h0 + lane + 32];
#else
  const float wh0 = weight[h0 + lane];
  const float wh1 = weight[h0 + lane + 32];
#endif
  float* op0 = out + (size_t)b * T * H + h0 + lane;
  float* op1 = op0 + 32;

  // Prologue: two chunks of prefetch in flight.
  issue_chunk(xbase, 0,      mylds[0], lane);
  issue_chunk(xbase, TC,     mylds[1], lane);

  for (int c = 0; c < NCHUNK; ++c) {
    if (c + 2 < NCHUNK) {
      issue_chunk(xbase, (c + 2) * TC, mylds[(c + 2) % NSTAGE], lane);
      WAIT_ASYNC(32);   // chunks c+1, c+2 (16 ops each) may remain in flight
    } else if (c + 1 < NCHUNK) {
      WAIT_ASYNC(16);   // only chunk c+1 still in flight
    } else {
      WAIT_ASYNC(0);    // drain
    }
    // This wave consumes only its own in-order async loads: no barrier.

    float (*bi)[WCOLS] = mylds[c % NSTAGE][0];
    float (*bf)[WCOLS] = mylds[c % NSTAGE][1];
#pragma unroll
    for (int t = 0; t < TC; ++t) {
      const float xin0 = bi[t][lane];
      const float xin1 = bi[t][lane + 32];
      const float xf0  = bf[t][lane];
      const float xf1  = bf[t][lane + 32];
      step(s0, xin0, xf0, wh0);   // two interleaved dependent chains
      step(s1, xin1, xf1, wh1);   // -> 2-way ILP on the critical path
      const size_t row = (size_t)(c * TC + t) * H;
      op0[row] = s0;              // fire-and-forget coalesced stores
      op1[row] = s1;
    }
  }
}

extern "C" void kernel_launch(void* const* d_in, const int* in_sizes, int n_in,
                              void* d_out, int out_size, void* d_ws, size_t ws_size,
                              hipStream_t stream) {
  const float* x      = (const float*)d_in[0];  // (8, 4096, 2048) f32
  const float* state  = (const float*)d_in[1];  // (8, 1024) f32
  const float* weight = (const float*)d_in[2];  // (1024,) f32
  float* out = (float*)d_out;                   // (8, 4096, 1024) f32

  dim3 grid(H / BH, B, 1);   // 8 x 8 = 64 blocks -> 64 WGPs
  dim3 block(64, 1, 1);      // 2 wave32 per block, barrier-free
  mrnn_scan_kernel<<<grid, block, 0, stream>>>(x, state, weight, out);
}